// HMM_11278584119744
// MI455X (gfx1250) — compile-verified
//
#include <hip/hip_runtime.h>

#define Hh      1024
#define VOCAB   32000
#define BATCH   32
#define SEQN    1024
#define NBLK    32          // persistent workgroups, one per WGP
#define ROWS    32          // alpha rows per workgroup (LDS-resident)
#define TPB     128         // 4 waves (wave32): 2x2 tiles of 16x16 => 32x32 block
#define LSTRIDE 1028        // padded LDS row stride (floats) for A rows / P columns

#define LOG2E 1.44269504088896340736f
#define LN2   0.69314718055994530942f

typedef float v2f __attribute__((ext_vector_type(2)));
typedef float v8f __attribute__((ext_vector_type(8)));

// LDS layout (in floats)
#define A_OFF   0
#define P_OFF   (ROWS * LSTRIDE)               // 32896 : P column-major, 32 cols
#define M_OFF   (P_OFF + BATCH * LSTRIDE)      // 65792 : 32 column maxes
#define R_OFF   (M_OFF + 32)                   // 65824 : final-reduce scratch
#define SMEM_FLOATS (R_OFF + 128)              // 65952 floats = 263808 B (< 320 KB)

__device__ __forceinline__ void grid_sync(unsigned* ctr, unsigned target) {
  __syncthreads();
  if (threadIdx.x == 0) {
    __threadfence();   // make this block's y_new visible at agent scope
    __hip_atomic_fetch_add(ctr, 1u, __ATOMIC_RELEASE, __HIP_MEMORY_SCOPE_AGENT);
    while (__hip_atomic_load(ctr, __ATOMIC_RELAXED, __HIP_MEMORY_SCOPE_AGENT) < target)
      __builtin_amdgcn_s_sleep(1);
    __threadfence();   // acquire: invalidate stale lines before re-reading y
  }
  __syncthreads();
}

extern "C" __global__ void hmm_init_kernel(unsigned* ctr) { *ctr = 0u; }

extern "C" __global__ __launch_bounds__(TPB)
void hmm_scan_kernel(const float* __restrict__ alpha,
                     const float* __restrict__ beta,
                     const float* __restrict__ gamma,
                     const int*   __restrict__ ids,
                     float* __restrict__ out,
                     float* __restrict__ yb0,   // y buffers are COLUMN-major:
                     float* __restrict__ yb1,   //   y[col*Hh + row]
                     unsigned* __restrict__ ctr)
{
  extern __shared__ float smem[];
  float* sA = smem + A_OFF;
  float* sP = smem + P_OFF;   // column-major: sP[col*LSTRIDE + k]
  float* sM = smem + M_OFF;
  float* sR = smem + R_OFF;

  const int tid  = threadIdx.x;
  const int wg   = blockIdx.x;
  const int wave = tid >> 5;
  const int lane = tid & 31;
  const int h    = lane >> 4;        // lane half (WMMA layout)
  const int ln   = lane & 15;
  const int m0   = (wave & 1) * 16;  // M-tile offset within 32-row block
  const int n0   = (wave >> 1) * 16; // N-tile offset within 32 batch cols
  const int c    = tid & 31;         // batch column (y0 init / final)
  const int rb   = tid >> 5;         // row group 0..3  (y0 init / final)
  const int row0 = wg * ROWS;

  // ---- Stage this WGP's 32x1024 alpha slice into LDS once (resident all steps)
  for (int idx = tid; idx < ROWS * Hh; idx += TPB) {
    int r = idx >> 10;
    int k = idx & (Hh - 1);
    sA[r * LSTRIDE + k] = alpha[(size_t)(row0 + r) * Hh + k];
  }

  // ---- y0 = beta[:, ids[:, SEQ-1]] : this block writes its 32-row slice
  {
    int tok = ids[c * SEQN + (SEQN - 1)];
    #pragma unroll
    for (int i = 0; i < 8; ++i) {
      int g = row0 + rb * 8 + i;
      yb0[c * Hh + g] = beta[(size_t)g * VOCAB + tok];
    }
  }
  grid_sync(ctr, NBLK);

  for (int s = 0; s < SEQN - 1; ++s) {
    const float* cur = (s & 1) ? yb1 : yb0;
    float*       nxt = (s & 1) ? yb0 : yb1;
    const int tt = SEQN - 2 - s;

    // prefetch this step's scattered emission gathers (hidden under GEMM)
    {
      int colg = n0 + ln;
      int tok  = ids[colg * SEQN + tt];
      int rbase = row0 + m0 + 8 * h;
      #pragma unroll
      for (int v = 0; v < 8; ++v)
        __builtin_prefetch(&beta[(size_t)(rbase + v) * VOCAB + tok], 0, 3);
    }

    // fused pass: per column (8 per wave): read y once (coalesced b128),
    // wave-reduce the max, exp2-with-FMA in registers, store P column to LDS.
    #pragma unroll 2
    for (int i = 0; i < 8; ++i) {
      const int col = wave * 8 + i;
      const float4* src = (const float4*)(cur + (size_t)col * Hh);
      float4 yv[8];
      #pragma unroll
      for (int j = 0; j < 8; ++j) yv[j] = src[lane + 32 * j];

      float mx = fmaxf(fmaxf(yv[0].x, yv[0].y), fmaxf(yv[0].z, yv[0].w));
      #pragma unroll
      for (int j = 1; j < 8; ++j)
        mx = fmaxf(mx, fmaxf(fmaxf(yv[j].x, yv[j].y), fmaxf(yv[j].z, yv[j].w)));
      #pragma unroll
      for (int d = 16; d > 0; d >>= 1)
        mx = fmaxf(mx, __shfl_xor(mx, d, 32));

      if (lane == 0) sM[col] = mx;

      const float nml = -mx * LOG2E;   // p = 2^(y*log2e - m*log2e)
      float* dst = sP + col * LSTRIDE + lane * 4;
      #pragma unroll
      for (int j = 0; j < 8; ++j) {
        float4 p;
        p.x = __builtin_amdgcn_exp2f(fmaf(yv[j].x, LOG2E, nml));
        p.y = __builtin_amdgcn_exp2f(fmaf(yv[j].y, LOG2E, nml));
        p.z = __builtin_amdgcn_exp2f(fmaf(yv[j].z, LOG2E, nml));
        p.w = __builtin_amdgcn_exp2f(fmaf(yv[j].w, LOG2E, nml));
        *(float4*)(dst + 128 * j) = p;
      }
    }
    __syncthreads();

    // GEMM: Z(32x32) = A(32x1024) * P(1024x32) with fp32 WMMA 16x16x4.
    // Two independent accumulator chains; operands are contiguous K-pairs
    // -> fused ds_load_2addr_b64 (1 DS op per WMMA).
    v8f acc0 = {0.f, 0.f, 0.f, 0.f, 0.f, 0.f, 0.f, 0.f};
    v8f acc1 = {0.f, 0.f, 0.f, 0.f, 0.f, 0.f, 0.f, 0.f};
    const float* ap = sA + (m0 + ln) * LSTRIDE + 2 * h;   // A[M][k0+2h, k0+2h+1]
    const float* bp = sP + (n0 + ln) * LSTRIDE + 2 * h;   // P[k0+2h, k0+2h+1][n]
    #pragma unroll 4
    for (int it = 0; it < Hh / 8; ++it) {
      v2f a0 = *(const v2f*)ap;
      v2f b0 = *(const v2f*)bp;
      v2f a1 = *(const v2f*)(ap + 4);
      v2f b1 = *(const v2f*)(bp + 4);
      acc0 = __builtin_amdgcn_wmma_f32_16x16x4_f32(false, a0, false, b0,
                                                   (short)0, acc0, false, false);
      acc1 = __builtin_amdgcn_wmma_f32_16x16x4_f32(false, a1, false, b1,
                                                   (short)0, acc1, false, false);
      ap += 8;
      bp += 8;
    }
    v8f acc = acc0 + acc1;

    // epilogue: y_new = e_t + ln(Z) + m  (C layout: VGPR v -> row m0+8h+v,
    // rows contiguous in column-major y -> two b128 stores)
    {
      int colg = n0 + ln;
      int tok  = ids[colg * SEQN + tt];
      float mc = sM[colg];
      int rbase = row0 + m0 + 8 * h;
      float r[8];
      #pragma unroll
      for (int v = 0; v < 8; ++v) {
        float e = beta[(size_t)(rbase + v) * VOCAB + tok];
        r[v] = fmaf(__builtin_amdgcn_logf(acc[v]), LN2, e) + mc;
      }
      float4 lo = { r[0], r[1], r[2], r[3] };
      float4 hi = { r[4], r[5], r[6], r[7] };
      *(float4*)&nxt[(size_t)colg * Hh + rbase]     = lo;
      *(float4*)&nxt[(size_t)colg * Hh + rbase + 4] = hi;
    }

    grid_sync(ctr, (unsigned)NBLK * (unsigned)(s + 2));
  }

  // ---- out = log_matmul(gamma_exp, y_final): block 0 only, 1x32 result
  if (wg == 0) {
    const float* y = ((SEQN - 1) & 1) ? yb1 : yb0;  // 1023 steps -> yb1
    float mx = -__builtin_inff();
    for (int i = 0; i < Hh / 4; ++i)
      mx = fmaxf(mx, y[c * Hh + rb + 4 * i]);
    sR[rb * 32 + c] = mx;
    __syncthreads();
    if (tid < 32)
      sM[tid] = fmaxf(fmaxf(sR[tid], sR[32 + tid]),
                      fmaxf(sR[64 + tid], sR[96 + tid]));
    __syncthreads();
    float m = sM[c];
    float sum = 0.f;
    for (int i = 0; i < Hh / 4; ++i) {
      int k = rb + 4 * i;
      sum += gamma[k] * __builtin_amdgcn_exp2f((y[c * Hh + k] - m) * LOG2E);
    }
    sR[rb * 32 + c] = sum;
    __syncthreads();
    if (tid < 32)
      out[tid] = fmaf(__builtin_amdgcn_logf(sR[tid] + sR[32 + tid] +
                                            sR[64 + tid] + sR[96 + tid]),
                      LN2, sM[tid]);
  }
}

extern "C" void kernel_launch(void* const* d_in, const int* in_sizes, int n_in,
                              void* d_out, int out_size, void* d_ws, size_t ws_size,
                              hipStream_t stream) {
  (void)in_sizes; (void)n_in; (void)out_size; (void)ws_size;
  const float* alpha = (const float*)d_in[0];
  const float* beta  = (const float*)d_in[1];
  const float* gamma = (const float*)d_in[2];
  const int*   ids   = (const int*)d_in[3];
  float* out = (float*)d_out;

  char* ws = (char*)d_ws;
  float* yb0 = (float*)ws;
  float* yb1 = (float*)(ws + (size_t)Hh * BATCH * sizeof(float));
  unsigned* ctr = (unsigned*)(ws + 2u * (size_t)Hh * BATCH * sizeof(float));

  size_t smem = (size_t)SMEM_FLOATS * sizeof(float);
  hipFuncSetAttribute(reinterpret_cast<const void*>(hmm_scan_kernel),
                      hipFuncAttributeMaxDynamicSharedMemorySize, (int)smem);

  hipLaunchKernelGGL(hmm_init_kernel, dim3(1), dim3(1), 0, stream, ctr);
  hipLaunchKernelGGL(hmm_scan_kernel, dim3(NBLK), dim3(TPB), smem, stream,
                     alpha, beta, gamma, ids, out, yb0, yb1, ctr);
}